// PointPillarsScatter_15006615733725
// MI455X (gfx1250) — compile-verified
//
#include <hip/hip_runtime.h>
#include <cstdint>

// Problem constants (match reference)
#define Bb 8
#define Cc 64
#define Hh 496
#define Ww 432
#define HW (Hh * Ww)          // 214272  (divisible by 128)
#define Pp 200000

typedef uint32_t u32;
typedef __attribute__((ext_vector_type(4))) u32 u32x4;
typedef __attribute__((ext_vector_type(4))) int i32x4;
typedef __attribute__((ext_vector_type(8))) int i32x8;

#define TILE 128              // canvas slots per workgroup (4 waves of 32)
#define ROWSTRIDE 68          // dwords per LDS row: 64 data + 4 TDM pad
                              // -> row base = 272B * tid, 16B aligned (legal ds_load_b128),
                              //    b128 reads hit each bank exactly twice (optimal 2-cycle drain)

// ---------------------------------------------------------------------------
// Phase 1: map[b*HW + hw] = -1  (6.86 MB in d_ws; int4 stores, exact fit)
// ---------------------------------------------------------------------------
__global__ void k_init_map(i32x4* __restrict__ map4) {
    int t = blockIdx.x * blockDim.x + threadIdx.x;
    i32x4 v = {-1, -1, -1, -1};
    map4[t] = v;
}

// ---------------------------------------------------------------------------
// Phase 2: scatter pillar ids into the map (coalesced int4 coord reads,
// tiny scattered 4B writes into a 6.9MB region -> all L2-resident).
// Overwrite semantics: arbitrary winner on duplicates, same as reference.
// ---------------------------------------------------------------------------
__global__ void k_scatter_pid(const int4* __restrict__ coords, int* __restrict__ map) {
    int p = blockIdx.x * blockDim.x + threadIdx.x;
    if (p >= Pp) return;
    int4 c = coords[p];                       // (b, x, y, z)
    int idx = c.y * Ww + c.z;
    idx = min(max(idx, 0), HW - 1);
    map[c.x * HW + idx] = p;
}

// ---------------------------------------------------------------------------
// Phase 3: gather. Each wave DMAs its 32 feature rows into LDS via TDM
// gather-mode descriptors (8 x 32-bit row indices each), waits on
// TENSORcnt, then streams the transposed tile out with NT stores.
// Empty slots use row index == Pp (OOB) -> TDM zero-fills, so the whole
// output is produced without any separate memset. Rows are wave-private,
// so no workgroup barrier is needed -- only s_wait_tensorcnt.
// ---------------------------------------------------------------------------
__global__ void __launch_bounds__(TILE)
k_gather(const float* __restrict__ feat, const int* __restrict__ map,
         float* __restrict__ out) {
    __shared__ float sh[TILE * ROWSTRIDE];    // 34,816 B -> ~9 WGs/WGP vs 320KB

    const int tid  = threadIdx.x;
    const int b    = blockIdx.y;
    const int hw0  = blockIdx.x * TILE;

    // slot -> pillar id (coalesced); OOB index for empty slots => zero rows
    int pid  = map[b * HW + hw0 + tid];
    int gidx = (pid < 0) ? Pp : pid;

    const u32 wv       = (u32)__builtin_amdgcn_readfirstlane(tid >> 5);
    const u32 lds_base = (u32)(uintptr_t)(&sh[0]);
    const uint64_t ga  = (uint64_t)(uintptr_t)feat;
    const u32 ga_lo    = (u32)ga;
    const u32 ga_hi    = (u32)(ga >> 32);

    // ---- D# group1 (constant): data_size=4B, pad 4 dwords every 64 dwords,
    // tensor_dim0=64, tensor_dim1=Pp (OOB bound), tile_dim0=64, tile_dim1=8
    // valid indices, tensor_dim0_stride=64 elements.
    i32x8 g1;
    g1[0] = (int)((2u << 16) | (1u << 20) | (5u << 22) | (3u << 25));
    g1[1] = (int)(64u << 16);                         // tensor_dim0[15:0]
    g1[2] = (int)(((u32)(Pp & 0xFFFF)) << 16);        // dim0 hi=0 | tensor_dim1[15:0]
    g1[3] = (int)((64u << 16) | ((u32)(Pp >> 16)));   // tile_dim0 | tensor_dim1[31:16]
    g1[4] = 8;                                        // tile_dim1 = 8 gathered rows
    g1[5] = 64;                                       // tensor_dim0_stride
    g1[6] = 0;
    g1[7] = 0;

#pragma unroll
    for (int j = 0; j < 4; ++j) {
        // ---- D# group0: count=1, gather_mode=1, 32-bit indices, lds/global addr
        u32x4 g0;
        g0[0] = 0xC0000001u;
        g0[1] = lds_base + (wv * 32u + (u32)j * 8u) * (ROWSTRIDE * 4u);
        g0[2] = ga_lo;
        g0[3] = (ga_hi & 0x01FFFFFFu) | 0x80000000u;  // addr[56:32] | type=2

        // ---- groups 2/3: eight 32-bit row indices from this wave's lanes
        i32x4 g2, g3;
        g2[0] = __builtin_amdgcn_readlane(gidx, j * 8 + 0);
        g2[1] = __builtin_amdgcn_readlane(gidx, j * 8 + 1);
        g2[2] = __builtin_amdgcn_readlane(gidx, j * 8 + 2);
        g2[3] = __builtin_amdgcn_readlane(gidx, j * 8 + 3);
        g3[0] = __builtin_amdgcn_readlane(gidx, j * 8 + 4);
        g3[1] = __builtin_amdgcn_readlane(gidx, j * 8 + 5);
        g3[2] = __builtin_amdgcn_readlane(gidx, j * 8 + 6);
        g3[3] = __builtin_amdgcn_readlane(gidx, j * 8 + 7);

#if __clang_major__ >= 23
        i32x8 gx = {0, 0, 0, 0, 0, 0, 0, 0};
        __builtin_amdgcn_tensor_load_to_lds(g0, g1, g2, g3, gx, 0);
#else
        __builtin_amdgcn_tensor_load_to_lds(g0, g1, g2, g3, 0);
#endif
    }

    __builtin_amdgcn_s_wait_tensorcnt(0);
    asm volatile("" ::: "memory");   // TDM wrote LDS behind the compiler's back

    // Transposed streaming write-out: 16B-aligned ds_load_b128 (2-cycle,
    // bank-balanced); stores are fully coalesced 128B/wave, non-temporal
    // (output is 439MB write-once, larger than the 192MB L2).
    const float* row = &sh[tid * ROWSTRIDE];
    float* op = out + ((size_t)b * Cc) * (size_t)HW + (size_t)(hw0 + tid);
#pragma unroll 16
    for (int c = 0; c < Cc; ++c) {
        __builtin_nontemporal_store(row[c], op + (size_t)c * (size_t)HW);
    }
}

// ---------------------------------------------------------------------------
// Fallback path (only if d_ws is too small for the 6.9MB map): zero fill +
// direct scatter. Slower (write amplification) but self-contained.
// ---------------------------------------------------------------------------
__global__ void k_zero_out(float4* __restrict__ out4) {
    size_t t = (size_t)blockIdx.x * blockDim.x + threadIdx.x;
    out4[t] = make_float4(0.f, 0.f, 0.f, 0.f);
}

__global__ void k_direct_scatter(const float* __restrict__ feat,
                                 const int4* __restrict__ coords,
                                 float* __restrict__ out) {
    int p = blockIdx.x;          // pillar
    int c = threadIdx.x;         // channel
    int4 cd = coords[p];
    int idx = min(max(cd.y * Ww + cd.z, 0), HW - 1);
    out[((size_t)cd.x * Cc + c) * (size_t)HW + idx] = feat[(size_t)p * Cc + c];
}

// ---------------------------------------------------------------------------
extern "C" void kernel_launch(void* const* d_in, const int* in_sizes, int n_in,
                              void* d_out, int out_size, void* d_ws, size_t ws_size,
                              hipStream_t stream) {
    const float* feat  = (const float*)d_in[0];
    const int*   coord = (const int*)d_in[1];
    float*       out   = (float*)d_out;

    const size_t mapBytes = (size_t)Bb * HW * sizeof(int);   // 6.86 MB

    if (ws_size >= mapBytes) {
        int* map = (int*)d_ws;
        // Phase 1: map = -1  (1,714,176 ints -> 428,544 int4 -> 1674 blocks exact)
        k_init_map<<<(Bb * HW / 4) / 256, 256, 0, stream>>>((i32x4*)map);
        // Phase 2: scatter pillar ids
        k_scatter_pid<<<(Pp + 255) / 256, 256, 0, stream>>>((const int4*)coord, map);
        // Phase 3: TDM gather + transposed NT write-out
        dim3 grid(HW / TILE, Bb);
        k_gather<<<grid, TILE, 0, stream>>>(feat, map, out);
    } else {
        // Fallback: zero output (27,426,816 float4 -> 107,136 blocks exact),
        // then direct scatter.
        k_zero_out<<<(unsigned)((size_t)Bb * Cc * HW / 4 / 256), 256, 0, stream>>>((float4*)out);
        k_direct_scatter<<<Pp, Cc, 0, stream>>>(feat, (const int4*)coord, out);
    }
}